// SlaterPooling_38723425141371
// MI455X (gfx1250) — compile-verified
//
#include <hip/hip_runtime.h>

typedef __attribute__((ext_vector_type(2))) float v2f;
typedef __attribute__((ext_vector_type(8))) float v8f;
typedef __attribute__((ext_vector_type(4))) int   v4i;

#define NBATCH 1024
#define NCONFS 64
#define NMO    64
#define NSIZE  32      // nup == ndown == 32
#define LDA    33      // padded leading dim (odd -> conflict-free banks)
#define WPB    8       // waves per block (256 threads, wave32)

#if __has_builtin(__builtin_amdgcn_global_load_async_to_lds_b128)
#define HAVE_ASYNC_LDS 1
#endif

__device__ __forceinline__ void wait_asynccnt0() {
#if __has_builtin(__builtin_amdgcn_s_wait_asynccnt)
  __builtin_amdgcn_s_wait_asynccnt(0);
#else
  asm volatile("s_wait_asynccnt 0x0" ::: "memory");
#endif
}

// In-wave blocked LU (no pivoting) of a 32x32 matrix living in LDS (row-major,
// leading dim LDA). Returns det (valid in all lanes). Uses fp32 WMMA for the
// 16x16x16 Schur-complement update. Pivot reciprocals use raw v_rcp_f32 to
// shorten the serial dependence chain (this kernel is latency-bound).
__device__ __forceinline__ float lu_det_32(float* __restrict__ A, const int lane) {
  const int row = lane;

  // ---- 1) Panel factorization: columns 0..15, all 32 rows (lane == row) ----
  for (int k = 0; k < 16; ++k) {
    const float rpiv = __builtin_amdgcn_rcpf(A[k * LDA + k]);
    if (row > k) {
      const float l = A[row * LDA + k] * rpiv;
      A[row * LDA + k] = l;
      for (int j = k + 1; j < 16; ++j)
        A[row * LDA + j] -= l * A[k * LDA + j];
    }
  }

  // ---- 2) Triangular solve: U12 = L11^{-1} * A12 (rows 0..15, cols 16..31) ----
  // lane j (0..15) owns column 16+j; dependency is only in k.
  if (lane < 16) {
    const int j = 16 + lane;
    for (int k = 0; k < 15; ++k) {
      const float ukj = A[k * LDA + j];
      for (int i = k + 1; i < 16; ++i)
        A[i * LDA + j] -= A[i * LDA + k] * ukj;
    }
  }

  // ---- 3) Schur complement: A22 = A22 - L21 * U12 via V_WMMA_F32_16X16X4_F32 ----
  // Full EXEC here (reconverged). fp32 WMMA has no A/B negation -> negate the
  // A-fragment (L21) in VALU so D = (-L21)*U12 + A22 accumulates correctly.
  {
    const int n  = lane & 15;   // C/D column, and A-frag row (M)
    const int hi = lane >> 4;   // half-wave select
    const int mb = hi * 8;      // C/D row base per ISA §7.12.2

    v8f acc;
#pragma unroll
    for (int v = 0; v < 8; ++v)
      acc[v] = A[(16 + mb + v) * LDA + 16 + n];

#pragma unroll
    for (int kb = 0; kb < 16; kb += 4) {
      const int kk = kb + hi * 2;          // K pair handled by this half-wave
      v2f afrag, bfrag;
      // A-frag: L21 is rows 16..31, cols kk..kk+1 (negated)
      afrag.x = -A[(16 + n) * LDA + kk];
      afrag.y = -A[(16 + n) * LDA + kk + 1];
      // B-frag: U12 rows kk..kk+1, col 16+n
      bfrag.x = A[kk * LDA + 16 + n];
      bfrag.y = A[(kk + 1) * LDA + 16 + n];
      acc = __builtin_amdgcn_wmma_f32_16x16x4_f32(
          /*neg_a=*/false, afrag, /*neg_b=*/false, bfrag,
          /*c_mod=*/(short)0, acc, /*reuse_a=*/false, /*reuse_b=*/false);
    }

#pragma unroll
    for (int v = 0; v < 8; ++v)
      A[(16 + mb + v) * LDA + 16 + n] = acc[v];
  }

  // ---- 4) Unblocked LU on trailing 16x16 (rows/cols 16..31), lane = row-16 ----
  if (lane < 16) {
    const int r2 = 16 + lane;
    for (int k = 16; k < 32; ++k) {
      const float rpiv = __builtin_amdgcn_rcpf(A[k * LDA + k]);
      if (r2 > k) {
        const float l = A[r2 * LDA + k] * rpiv;
        A[r2 * LDA + k] = l;
        for (int j = k + 1; j < 32; ++j)
          A[r2 * LDA + j] -= l * A[k * LDA + j];
      }
    }
  }

  // ---- 5) det = product of diagonal, wave32 butterfly reduction ----
  float d = A[lane * LDA + lane];
#pragma unroll
  for (int off = 16; off > 0; off >>= 1)
    d *= __shfl_xor(d, off, 32);
  return d;
}

__global__ __launch_bounds__(256)
void slater_pooling_kernel(const float* __restrict__ mo,
                           const int* __restrict__ cup,
                           const int* __restrict__ cdown,
                           float* __restrict__ out) {
  __shared__ float moTile[NMO * NMO];       // 16 KB: mo[b], 64 electrons x 64 orbitals
  __shared__ float Amat[WPB][NSIZE * LDA];  // per-wave 32x33 working matrix

  const int b    = blockIdx.x;
  const int tid  = threadIdx.x;
  const int lane = tid & 31;
  const int wave = tid >> 5;

  // Stage mo[b] into LDS once per block. Prefer the CDNA5 async DMA path
  // (GLOBAL_LOAD_ASYNC_TO_LDS_B128, ASYNCcnt-tracked, no VGPR round-trip).
  {
    const float4* src = (const float4*)(mo + (size_t)b * NMO * NMO);
    float4*       dst = (float4*)moTile;
#if defined(HAVE_ASYNC_LDS)
#pragma unroll
    for (int t = 0; t < 4; ++t) {
      __builtin_amdgcn_global_load_async_to_lds_b128(
          (v4i*)(src + tid + t * 256),
          (v4i*)(dst + tid + t * 256),
          /*offset=*/0, /*cpol=*/0);
    }
    wait_asynccnt0();   // drain this wave's async DMA before the barrier
#else
#pragma unroll
    for (int t = 0; t < 4; ++t)
      dst[tid + t * 256] = src[tid + t * 256];
#endif
  }
  __syncthreads();

  float* A = Amat[wave];

  for (int t = 0; t < NCONFS / WPB; ++t) {
    const int c = t * WPB + wave;

    // Gather spin-up Slater matrix: A[i][lane] = mo[b][i][cup[c][lane]]
    {
      const int col = cup[c * NSIZE + lane];
      for (int i = 0; i < NSIZE; ++i)
        A[i * LDA + lane] = moTile[i * NMO + col];
    }
    const float det_u = lu_det_32(A, lane);

    // Gather spin-down Slater matrix: A[i][lane] = mo[b][32+i][cdown[c][lane]]
    {
      const int col = cdown[c * NSIZE + lane];
      for (int i = 0; i < NSIZE; ++i)
        A[i * LDA + lane] = moTile[(NSIZE + i) * NMO + col];
    }
    const float det_d = lu_det_32(A, lane);

    if (lane == 0)
      out[(size_t)b * NCONFS + c] = det_u * det_d;
  }
}

extern "C" void kernel_launch(void* const* d_in, const int* in_sizes, int n_in,
                              void* d_out, int out_size, void* d_ws, size_t ws_size,
                              hipStream_t stream) {
  const float* mo    = (const float*)d_in[0];
  const int*   cup   = (const int*)d_in[1];
  const int*   cdown = (const int*)d_in[2];
  float*       out   = (float*)d_out;

  slater_pooling_kernel<<<dim3(NBATCH), dim3(256), 0, stream>>>(mo, cup, cdown, out);
}